// SimpleMessagePassing_14929306321609
// MI455X (gfx1250) — compile-verified
//
#include <hip/hip_runtime.h>
#include <math.h>

typedef __attribute__((ext_vector_type(2))) float v2f;
typedef __attribute__((ext_vector_type(8))) float v8f;

#define IN_CH   128
#define OUT_CH  64
#define HEADS   4
#define QKV_CH  256   // OUT_CH * HEADS

// ---------------------------------------------------------------------------
// Projection GEMM: Q/K/V = x @ W   (M = n_nodes, K = 128, N = 256)
// blockDim = 256 (8 wave32), grid = (M/16, 6).
// Wave w of block (mt, by) owns tile t = by*8 + w in {matrix 0..2} x {N-tile 0..15}.
// One wave owns one 16x16 fp32 C tile; K-loop steps of 4 via
// V_WMMA_F32_16X16X4_F32 (full fp32: reference is fp32 and the GEMMs are not
// the bottleneck, so no precision trade).
// ---------------------------------------------------------------------------
__global__ void __launch_bounds__(256)
proj_wmma(const float* __restrict__ x,
          const float* __restrict__ Wq,
          const float* __restrict__ Wk,
          const float* __restrict__ Wv,
          float* __restrict__ Q,
          float* __restrict__ Kf,
          float* __restrict__ Vf,
          int n_nodes)
{
    const int mt   = blockIdx.x;
    const int wave = threadIdx.x >> 5;
    const int t    = blockIdx.y * 8 + wave;   // 0..47
    const int mat  = t >> 4;                  // 0=Q, 1=K, 2=V
    const int nt   = t & 15;                  // N tile

    const float* __restrict__ W = (mat == 0) ? Wq : (mat == 1) ? Wk : Wv;
    float* __restrict__ O       = (mat == 0) ? Q  : (mat == 1) ? Kf : Vf;

    const int lane  = threadIdx.x & 31;
    const int l15   = lane & 15;
    const int khalf = lane >> 4;                 // K pair {0,1} vs {2,3}
    int row = mt * 16 + l15;                     // A-matrix M index for this lane
    if (row >= n_nodes) row = n_nodes - 1;       // clamp (N is divisible by 16)
    const int col = nt * 16 + l15;               // B/C/D N index for this lane

    const float* __restrict__ arow = x + (size_t)row * IN_CH;

    // pull the L2-resident weight panel toward this WGP (global_prefetch_b8)
    __builtin_prefetch(&W[col], 0, 3);
    __builtin_prefetch(&W[64 * QKV_CH + col], 0, 3);

    v8f acc = {};
    #pragma unroll 8
    for (int k0 = 0; k0 < IN_CH; k0 += 4) {
        const int ka = k0 + 2 * khalf;
        v2f a = *(const v2f*)(arow + ka);        // A: lane-half picks K high bit
        v2f b;                                   // B: VGPR picks K low bit
        b.x = W[(size_t)(ka + 0) * QKV_CH + col];
        b.y = W[(size_t)(ka + 1) * QKV_CH + col];
        acc = __builtin_amdgcn_wmma_f32_16x16x4_f32(false, a, false, b,
                                                    (short)0, acc, false, false);
    }

    // C/D layout: VGPR r -> M = r + 8*khalf. Branchless full-tile fast path.
    const int mbase = mt * 16;
    float* __restrict__ orow = O + (size_t)(mbase + 8 * khalf) * QKV_CH + col;
    if (mbase + 16 <= n_nodes) {
        #pragma unroll
        for (int r = 0; r < 8; ++r)
            orow[(size_t)r * QKV_CH] = acc[r];
    } else {
        #pragma unroll
        for (int r = 0; r < 8; ++r) {
            const int m = mbase + r + 8 * khalf;
            if (m < n_nodes) O[(size_t)m * QKV_CH + col] = acc[r];
        }
    }
}

// ---------------------------------------------------------------------------
// Per-edge attention logit: one wave32 per edge, 8 channels/lane.
// attn[e][h] = leaky_relu( dot(Q[tgt], K[src])_h / 8 + ew[e]*We[h] )
// Q+K (102 MB) are L2-resident on MI455X (192 MB L2): random per-edge gathers
// stay on-chip instead of hitting HBM.
// ---------------------------------------------------------------------------
__global__ void __launch_bounds__(256)
edge_attn(const float* __restrict__ Q,
          const float* __restrict__ Kf,
          const int* __restrict__ src,
          const int* __restrict__ tgt,
          const float* __restrict__ ew,
          const float* __restrict__ We,
          float* __restrict__ attn,
          int E)
{
    const int gid  = blockIdx.x * blockDim.x + threadIdx.x;
    const int e    = gid >> 5;
    const int lane = threadIdx.x & 31;
    if (e >= E) return;

    const int s = src[e];
    const int t = tgt[e];
    const float4* __restrict__ q4 = (const float4*)(Q  + (size_t)t * QKV_CH);
    const float4* __restrict__ k4 = (const float4*)(Kf + (size_t)s * QKV_CH);

    float4 qa = q4[lane * 2 + 0], qb = q4[lane * 2 + 1];
    float4 ka = k4[lane * 2 + 0], kb = k4[lane * 2 + 1];
    float p = qa.x * ka.x + qa.y * ka.y + qa.z * ka.z + qa.w * ka.w
            + qb.x * kb.x + qb.y * kb.y + qb.z * kb.z + qb.w * kb.w;

    // reduce across the 8 lanes sharing a head
    p += __shfl_xor(p, 1, 32);
    p += __shfl_xor(p, 2, 32);
    p += __shfl_xor(p, 4, 32);

    const int h = lane >> 3;
    if ((lane & 7) == 0) {
        float a = p * 0.125f + ew[e] * We[h];    // 1/sqrt(64) = 0.125
        a = (a > 0.0f) ? a : 0.2f * a;           // leaky_relu(0.2)
        attn[(size_t)e * HEADS + h] = a;
    }
}

// ---------------------------------------------------------------------------
// Deterministic global softmax reductions (per head): fixed-grid partials +
// single-block final, tree reductions in LDS.
// ---------------------------------------------------------------------------
__global__ void __launch_bounds__(256)
softmax_max_partial(const float4* __restrict__ attn4, int E,
                    float4* __restrict__ pmax)
{
    __shared__ float4 sm[256];
    float4 m = make_float4(-3.4e38f, -3.4e38f, -3.4e38f, -3.4e38f);
    for (int e = blockIdx.x * blockDim.x + threadIdx.x; e < E;
         e += gridDim.x * blockDim.x) {
        float4 v = attn4[e];
        m.x = fmaxf(m.x, v.x); m.y = fmaxf(m.y, v.y);
        m.z = fmaxf(m.z, v.z); m.w = fmaxf(m.w, v.w);
    }
    sm[threadIdx.x] = m;
    __syncthreads();
    for (int s = 128; s > 0; s >>= 1) {
        if (threadIdx.x < s) {
            float4 o = sm[threadIdx.x + s];
            float4 c = sm[threadIdx.x];
            c.x = fmaxf(c.x, o.x); c.y = fmaxf(c.y, o.y);
            c.z = fmaxf(c.z, o.z); c.w = fmaxf(c.w, o.w);
            sm[threadIdx.x] = c;
        }
        __syncthreads();
    }
    if (threadIdx.x == 0) pmax[blockIdx.x] = sm[0];
}

__global__ void __launch_bounds__(256)
softmax_max_final(const float4* __restrict__ pmax, float4* __restrict__ gmax)
{
    __shared__ float4 sm[256];
    sm[threadIdx.x] = pmax[threadIdx.x];
    __syncthreads();
    for (int s = 128; s > 0; s >>= 1) {
        if (threadIdx.x < s) {
            float4 o = sm[threadIdx.x + s];
            float4 c = sm[threadIdx.x];
            c.x = fmaxf(c.x, o.x); c.y = fmaxf(c.y, o.y);
            c.z = fmaxf(c.z, o.z); c.w = fmaxf(c.w, o.w);
            sm[threadIdx.x] = c;
        }
        __syncthreads();
    }
    if (threadIdx.x == 0) *gmax = sm[0];
}

__global__ void __launch_bounds__(256)
softmax_sum_partial(const float4* __restrict__ attn4, int E,
                    const float4* __restrict__ gmax,
                    float4* __restrict__ psum)
{
    __shared__ float4 sm[256];
    const float4 g = *gmax;
    float4 acc = make_float4(0.f, 0.f, 0.f, 0.f);
    for (int e = blockIdx.x * blockDim.x + threadIdx.x; e < E;
         e += gridDim.x * blockDim.x) {
        float4 v = attn4[e];
        acc.x += expf(v.x - g.x); acc.y += expf(v.y - g.y);
        acc.z += expf(v.z - g.z); acc.w += expf(v.w - g.w);
    }
    sm[threadIdx.x] = acc;
    __syncthreads();
    for (int s = 128; s > 0; s >>= 1) {
        if (threadIdx.x < s) {
            float4 o = sm[threadIdx.x + s];
            float4 c = sm[threadIdx.x];
            c.x += o.x; c.y += o.y; c.z += o.z; c.w += o.w;
            sm[threadIdx.x] = c;
        }
        __syncthreads();
    }
    if (threadIdx.x == 0) psum[blockIdx.x] = sm[0];
}

__global__ void __launch_bounds__(256)
softmax_sum_final(const float4* __restrict__ psum, float4* __restrict__ gsum)
{
    __shared__ float4 sm[256];
    sm[threadIdx.x] = psum[threadIdx.x];
    __syncthreads();
    for (int s = 128; s > 0; s >>= 1) {
        if (threadIdx.x < s) {
            float4 o = sm[threadIdx.x + s];
            float4 c = sm[threadIdx.x];
            c.x += o.x; c.y += o.y; c.z += o.z; c.w += o.w;
            sm[threadIdx.x] = c;
        }
        __syncthreads();
    }
    if (threadIdx.x == 0) *gsum = sm[0];
}

// ---------------------------------------------------------------------------
// Weighted scatter: acc[tgt[e]] += softmax(attn)[e,h] * V[src[e]]
// One wave32 per edge; global_atomic_add_f32 into the L2-resident 51 MB acc.
// ---------------------------------------------------------------------------
__global__ void __launch_bounds__(256)
scatter_av(const float* __restrict__ Vf,
           const float* __restrict__ attn,
           const int* __restrict__ src,
           const int* __restrict__ tgt,
           const float4* __restrict__ gmax,
           const float4* __restrict__ gsum,
           float* __restrict__ acc,
           int E)
{
    const int gid  = blockIdx.x * blockDim.x + threadIdx.x;
    const int e    = gid >> 5;
    const int lane = threadIdx.x & 31;
    if (e >= E) return;

    const int h = lane >> 3;
    const int s = src[e];
    const int t = tgt[e];
    const float* gm = (const float*)gmax;
    const float* gs = (const float*)gsum;
    const float a = attn[(size_t)e * HEADS + h];
    const float w = expf(a - gm[h]) / gs[h];

    const float4* __restrict__ v4 = (const float4*)(Vf + (size_t)s * QKV_CH);
    float4 va = v4[lane * 2 + 0], vb = v4[lane * 2 + 1];
    float* o = acc + (size_t)t * QKV_CH + lane * 8;
    atomicAdd(o + 0, w * va.x); atomicAdd(o + 1, w * va.y);
    atomicAdd(o + 2, w * va.z); atomicAdd(o + 3, w * va.w);
    atomicAdd(o + 4, w * vb.x); atomicAdd(o + 5, w * vb.y);
    atomicAdd(o + 6, w * vb.z); atomicAdd(o + 7, w * vb.w);
}

__global__ void __launch_bounds__(256)
zero_f4(float4* __restrict__ p, size_t n4)
{
    size_t i = (size_t)blockIdx.x * blockDim.x + threadIdx.x;
    const size_t stride = (size_t)gridDim.x * blockDim.x;
    const float4 z = make_float4(0.f, 0.f, 0.f, 0.f);
    for (; i < n4; i += stride) p[i] = z;
}

// ---------------------------------------------------------------------------
// Output GEMM: out = acc[N,256] @ Wout[256,64] + b  (fp32 WMMA)
// blockDim = 128 (4 wave32, one per N-tile), grid = (M/16)
// ---------------------------------------------------------------------------
__global__ void __launch_bounds__(128)
out_wmma(const float* __restrict__ A,      // acc, ld = 256
         const float* __restrict__ Wout,   // 256 x 64
         const float* __restrict__ bias,   // 64
         float* __restrict__ out,          // N x 64
         int n_nodes)
{
    const int mt    = blockIdx.x;
    const int nt    = threadIdx.x >> 5;    // 0..3
    const int lane  = threadIdx.x & 31;
    const int l15   = lane & 15;
    const int khalf = lane >> 4;
    int row = mt * 16 + l15;
    if (row >= n_nodes) row = n_nodes - 1;
    const int col = nt * 16 + l15;

    const float* __restrict__ arow = A + (size_t)row * QKV_CH;

    __builtin_prefetch(&Wout[col], 0, 3);
    __builtin_prefetch(&Wout[128 * OUT_CH + col], 0, 3);

    v8f accv = {};
    #pragma unroll 8
    for (int k0 = 0; k0 < QKV_CH; k0 += 4) {
        const int ka = k0 + 2 * khalf;
        v2f a = *(const v2f*)(arow + ka);
        v2f b;
        b.x = Wout[(size_t)(ka + 0) * OUT_CH + col];
        b.y = Wout[(size_t)(ka + 1) * OUT_CH + col];
        accv = __builtin_amdgcn_wmma_f32_16x16x4_f32(false, a, false, b,
                                                     (short)0, accv, false, false);
    }

    const float bv = bias[col];
    const int mbase = mt * 16;
    float* __restrict__ orow = out + (size_t)(mbase + 8 * khalf) * OUT_CH + col;
    if (mbase + 16 <= n_nodes) {
        #pragma unroll
        for (int r = 0; r < 8; ++r)
            orow[(size_t)r * OUT_CH] = accv[r] + bv;
    } else {
        #pragma unroll
        for (int r = 0; r < 8; ++r) {
            const int m = mbase + r + 8 * khalf;
            if (m < n_nodes) out[(size_t)m * OUT_CH + col] = accv[r] + bv;
        }
    }
}

// ---------------------------------------------------------------------------
// Host-side launch. Workspace layout (bytes):
//   [0)        Q   : N*256 f32   (reused as scatter accumulator)
//   [qkv)      K   : N*256 f32
//   [2*qkv)    V   : N*256 f32
//   [3*qkv)    attn: E*4 f32
//   [+attn)    pmax: 256 float4, psum: 256 float4, gmax, gsum
// ---------------------------------------------------------------------------
extern "C" void kernel_launch(void* const* d_in, const int* in_sizes, int n_in,
                              void* d_out, int out_size, void* d_ws, size_t ws_size,
                              hipStream_t stream)
{
    const float* x    = (const float*)d_in[0];
    const int*   ei   = (const int*)  d_in[1];
    const float* ew   = (const float*)d_in[2];
    const float* Wq   = (const float*)d_in[3];
    const float* Wk   = (const float*)d_in[4];
    const float* Wv   = (const float*)d_in[5];
    const float* We   = (const float*)d_in[6];
    const float* Wout = (const float*)d_in[7];
    const float* bout = (const float*)d_in[8];
    float* out = (float*)d_out;

    const int n_nodes = in_sizes[0] / IN_CH;
    const int E       = in_sizes[2];          // edge_weights is [E,1]
    const int* src = ei;
    const int* tgt = ei + E;

    char* ws = (char*)d_ws;
    const size_t qkv_bytes  = (size_t)n_nodes * QKV_CH * sizeof(float);
    const size_t attn_bytes = (size_t)E * HEADS * sizeof(float);
    float*  Q    = (float*)(ws);
    float*  Kf   = (float*)(ws + qkv_bytes);
    float*  Vf   = (float*)(ws + 2 * qkv_bytes);
    float*  attn = (float*)(ws + 3 * qkv_bytes);
    float4* pmax = (float4*)(ws + 3 * qkv_bytes + attn_bytes);
    float4* psum = pmax + 256;
    float4* gmax = psum + 256;
    float4* gsum = gmax + 1;

    const int mtiles = (n_nodes + 15) / 16;

    // 1) node-level projections (10x fewer FLOPs than the reference's
    //    per-edge projections; identical math by linearity)
    proj_wmma<<<dim3(mtiles, 6), 256, 0, stream>>>(x, Wq, Wk, Wv, Q, Kf, Vf, n_nodes);

    // 2) per-edge attention logits
    const int ewarp_blocks = (E + 7) / 8;     // 8 wave32 per 256-thread block
    edge_attn<<<ewarp_blocks, 256, 0, stream>>>(Q, Kf, src, tgt, ew, We, attn, E);

    // 3) deterministic global softmax stats (per head)
    softmax_max_partial<<<256, 256, 0, stream>>>((const float4*)attn, E, pmax);
    softmax_max_final  <<<1,   256, 0, stream>>>(pmax, gmax);
    softmax_sum_partial<<<256, 256, 0, stream>>>((const float4*)attn, E, gmax, psum);
    softmax_sum_final  <<<1,   256, 0, stream>>>(psum, gsum);

    // 4) weighted scatter-add into accumulator (reuses Q's storage)
    float* acc = Q;
    const size_t n4 = (size_t)n_nodes * QKV_CH / 4;
    zero_f4<<<2048, 256, 0, stream>>>((float4*)acc, n4);
    scatter_av<<<ewarp_blocks, 256, 0, stream>>>(Vf, attn, src, tgt, gmax, gsum, acc, E);

    // 5) final projection + bias
    out_wmma<<<mtiles, 128, 0, stream>>>(acc, Wout, bout, out, n_nodes);
}